// PointsRasterizer_67697274520442
// MI455X (gfx1250) — compile-verified
//
#include <hip/hip_runtime.h>
#include <hip/hip_bf16.h>

typedef float v2f __attribute__((ext_vector_type(2)));
typedef float v8f __attribute__((ext_vector_type(8)));

#define HB 2
#define NP 32768
#define HH 256
#define WW 256
#define KK 8
#define RAD 0.01f
#define RR2 (RAD * RAD)
#define BIG 1e10f
#define CAP 1024
#define NTILE 16
#define BHWK (HB * HH * WW * KK)

// Async global->LDS (CDNA5): use if the toolchain declares the builtins.
#if defined(__gfx1250__) && __has_builtin(__builtin_amdgcn_global_load_async_to_lds_b128) && \
    __has_builtin(__builtin_amdgcn_s_wait_asynccnt)
#define USE_ASYNC_LDS 1
#else
#define USE_ASYNC_LDS 0
#endif

#if USE_ASYNC_LDS
// Builtin signature (from hipcc diagnostic): param0 = addrspace(1) int __vector(4)*,
// param1 = addrspace(3) int __vector(4)*, then two integer immediates.
typedef int v4i_vs __attribute__((__vector_size__(16)));
typedef __attribute__((address_space(1))) v4i_vs g_v4i;
typedef __attribute__((address_space(3))) v4i_vs l_v4i;
#endif

// ---------------------------------------------------------------------------
// 1) Transform: hom[16x4] x [FP | WV | 0][4x16] via V_WMMA_F32_16X16X4_F32.
//    D[m][0..3] = proj(point m), D[m][4..7] = view(point m).
//    Output: float4 (ndc_x, ndc_y, view_z, idx-bits) for aligned b128 traffic.
// ---------------------------------------------------------------------------
__global__ void xform_kernel(const float* __restrict__ pts_in,
                             const float* __restrict__ fullproj,
                             const float* __restrict__ worldview,
                             float4* __restrict__ pts_out) {
    __shared__ float lds[8 * 256];
    const int lane  = threadIdx.x & 31;
    const int wave  = threadIdx.x >> 5;
    const int group = blockIdx.x * 8 + wave;   // one 16-point group per wave
    const int gpb   = NP / 16;                 // groups per batch
    const int b     = group / gpb;
    const int g     = group - b * gpb;
    const int m     = lane & 15;
    const int pbase = g * 16;

    // A-matrix (16x4 f32): lanes 0-15 -> K=0,1 ; lanes 16-31 -> K=2,3
    const float* P = pts_in + (size_t)(b * NP + pbase + m) * 3;
    v2f a;
    if (lane < 16) { a.x = P[0]; a.y = P[1]; }
    else           { a.x = P[2]; a.y = 1.0f; }

    // B-matrix (4x16 f32): cols 0-3 = full_proj, cols 4-7 = world_view, rest 0
    const float* M0 = fullproj  + b * 16;
    const float* M1 = worldview + b * 16;
    const int n  = lane & 15;
    const int k0 = (lane >> 4) << 1;
    v2f bm;
    bm.x = (n < 4) ? M0[k0 * 4 + n] : ((n < 8) ? M1[k0 * 4 + (n - 4)] : 0.0f);
    bm.y = (n < 4) ? M0[(k0 + 1) * 4 + n] : ((n < 8) ? M1[(k0 + 1) * 4 + (n - 4)] : 0.0f);

    v8f c = {};
    v8f d = __builtin_amdgcn_wmma_f32_16x16x4_f32(
        /*neg_a=*/false, a, /*neg_b=*/false, bm,
        /*c_mod=*/(short)0, c, /*reuse_a=*/false, /*reuse_b=*/false);

    // D layout: VGPR r, lanes 0-15 -> D[r][lane]; lanes 16-31 -> D[8+r][lane-16]
    float* L = lds + wave * 256;
#pragma unroll
    for (int r = 0; r < 8; ++r) L[r * 32 + lane] = d[r];
    __syncthreads();

    if (lane < 16) {
        const int rr = m & 7;
        const int lo = (m < 8) ? 0 : 16;
        const float prx = L[rr * 32 + lo + 0];
        const float pry = L[rr * 32 + lo + 1];
        const float prw = L[rr * 32 + lo + 3];
        const float vz  = L[rr * 32 + lo + 6];
        const float vw  = L[rr * 32 + lo + 7];
        const float inv = 1.0f / prw;
        const int p = pbase + m;               // per-batch point index
        pts_out[(size_t)b * NP + p] =
            make_float4(prx * inv, pry * inv, vz / vw, __int_as_float(p));
    }
}

// ---------------------------------------------------------------------------
// 2) Binning: each point covers <= 2x2 of the 16x16 pixel tiles (2R << tile).
// ---------------------------------------------------------------------------
__global__ void zero_kernel(int* __restrict__ cnt) {
    if (threadIdx.x < HB * NTILE * NTILE) cnt[threadIdx.x] = 0;
}

__global__ void bin_kernel(const float4* __restrict__ pts,
                           int* __restrict__ cnt, int* __restrict__ list) {
    const int i = blockIdx.x * blockDim.x + threadIdx.x;
    if (i >= HB * NP) return;
    const int b = i / NP;
    const int p = i - b * NP;
    const float4 s = pts[i];                 // single global_load_b128
    if (!(s.z > 0.0f)) return;
    // pixel centers: (j + 0.5)/128 - 1 ; per-axis |center - pt| <= RAD superset
    int jx0 = (int)ceilf ((s.x - RAD + 1.0f) * (0.5f * WW) - 0.5f);
    int jx1 = (int)floorf((s.x + RAD + 1.0f) * (0.5f * WW) - 0.5f);
    int jy0 = (int)ceilf ((s.y - RAD + 1.0f) * (0.5f * HH) - 0.5f);
    int jy1 = (int)floorf((s.y + RAD + 1.0f) * (0.5f * HH) - 0.5f);
    jx0 = max(jx0, 0);      jy0 = max(jy0, 0);
    jx1 = min(jx1, WW - 1); jy1 = min(jy1, HH - 1);
    if (jx0 > jx1 || jy0 > jy1) return;
    const int tx0 = jx0 >> 4, tx1 = jx1 >> 4;
    const int ty0 = jy0 >> 4, ty1 = jy1 >> 4;
    for (int ty = ty0; ty <= ty1; ++ty)
        for (int tx = tx0; tx <= tx1; ++tx) {
            const int t = b * (NTILE * NTILE) + ty * NTILE + tx;
            const int slot = atomicAdd(&cnt[t], 1);
            if (slot < CAP) list[(size_t)t * CAP + slot] = p;
        }
}

// ---------------------------------------------------------------------------
// 3) Rasterize: one workgroup per tile (256 threads = 8 wave32 = 1 pixel/thread)
//    Stream tile point list through LDS (async global->LDS on CDNA5);
//    keep top-8 by (z, idx) in registers.
// ---------------------------------------------------------------------------
__device__ __forceinline__ void insert8(float z, int id, float d2,
                                        float (&bz)[KK], int (&bi)[KK], float (&bd)[KK]) {
    if (!((z < bz[KK - 1]) || (z == bz[KK - 1] && id < bi[KK - 1]))) return;
    bz[KK - 1] = z; bi[KK - 1] = id; bd[KK - 1] = d2;
#pragma unroll
    for (int k = KK - 1; k > 0; --k) {
        const bool sw = (bz[k] < bz[k - 1]) || (bz[k] == bz[k - 1] && bi[k] < bi[k - 1]);
        if (sw) {
            float tz = bz[k]; bz[k] = bz[k - 1]; bz[k - 1] = tz;
            int   ti = bi[k]; bi[k] = bi[k - 1]; bi[k - 1] = ti;
            float td = bd[k]; bd[k] = bd[k - 1]; bd[k - 1] = td;
        }
    }
}

__global__ void raster_kernel(const float4* __restrict__ pts,
                              const int* __restrict__ cnt,
                              const int* __restrict__ list,
                              int* __restrict__ idx_out,
                              float* __restrict__ z_out,
                              float* __restrict__ d2_out) {
    const int tile = blockIdx.x;            // 0..511
    const int b  = tile >> 8;
    const int t  = tile & 255;
    const int ty = t >> 4, tx = t & 15;
    const int r  = threadIdx.x >> 4, c = threadIdx.x & 15;
    const int py = ty * 16 + r, px = tx * 16 + c;
    const float cx = (2.0f * px + 1.0f) / (float)WW - 1.0f;
    const float cy = (2.0f * py + 1.0f) / (float)HH - 1.0f;

    const int n = min(cnt[tile], CAP);
    const int* L = list + (size_t)tile * CAP;
    const float4* P4 = pts + (size_t)b * NP;

    float bz[KK], bd[KK];
    int   bi[KK];
#pragma unroll
    for (int k = 0; k < KK; ++k) { bz[k] = BIG; bi[k] = 0x7fffffff; bd[k] = 0.0f; }

    __shared__ float4 sp[256];
    for (int base = 0; base < n; base += 256) {
        const int j = base + threadIdx.x;
        if (j < n) {
            const int pi = L[j];
#if USE_ASYNC_LDS
            // per-lane gather straight into LDS, tracked by ASYNCcnt
            __builtin_amdgcn_global_load_async_to_lds_b128(
                (g_v4i*)(uintptr_t)(P4 + pi),
                (l_v4i*)(unsigned int)(uintptr_t)(&sp[threadIdx.x]),
                0, 0);
#else
            sp[threadIdx.x] = P4[pi];       // b128 load + ds_store_b128
#endif
        }
        if (base + 256 < n)  // hint next chunk of the list (global_prefetch_b8)
            __builtin_prefetch(&L[base + 256 + threadIdx.x], 0, 1);
#if USE_ASYNC_LDS
        __builtin_amdgcn_s_wait_asynccnt(0);
#endif
        __syncthreads();
        const int m = min(256, n - base);
        for (int k = 0; k < m; ++k) {
            const float4 s = sp[k];
            const float dx = cx - s.x, dy = cy - s.y;
            const float d2 = dx * dx + dy * dy;
            if (d2 <= RR2 && s.z > 0.0f)
                insert8(s.z, __float_as_int(s.w), d2, bz, bi, bd);
        }
        __syncthreads();
    }

    const size_t o = ((size_t)((b * HH + py) * WW + px)) * KK;
#pragma unroll
    for (int k = 0; k < KK; ++k) {
        const bool empty = (bz[k] >= BIG);
        idx_out[o + k] = empty ? -1    : bi[k];
        z_out[o + k]   = empty ? -1.0f : bz[k];
        d2_out[o + k]  = empty ? -1.0f : bd[k];
    }
}

// ---------------------------------------------------------------------------
extern "C" void kernel_launch(void* const* d_in, const int* in_sizes, int n_in,
                              void* d_out, int out_size, void* d_ws, size_t ws_size,
                              hipStream_t stream) {
    const float* points    = (const float*)d_in[0];   // [2,32768,3]
    const float* fullproj  = (const float*)d_in[1];   // [2,4,4]
    const float* worldview = (const float*)d_in[2];   // [2,4,4]

    // Workspace layout (float4-aligned points first)
    float4* pts = (float4*)d_ws;                        // 2*32768 float4 (1 MB)
    int*    cnt = (int*)((float*)d_ws + (size_t)HB * NP * 4);  // 512 ints
    int*    lst = cnt + HB * NTILE * NTILE;             // 512*CAP ints (2 MB)

    // Outputs: (idx, zbuf, dists2) concatenated flat
    int*   idx_out = (int*)d_out;
    float* z_out   = (float*)d_out + BHWK;
    float* d2_out  = (float*)d_out + 2 * BHWK;

    hipLaunchKernelGGL(zero_kernel, dim3(1), dim3(512), 0, stream, cnt);
    hipLaunchKernelGGL(xform_kernel, dim3((HB * NP / 16) / 8), dim3(256), 0, stream,
                       points, fullproj, worldview, pts);
    hipLaunchKernelGGL(bin_kernel, dim3((HB * NP + 255) / 256), dim3(256), 0, stream,
                       pts, cnt, lst);
    hipLaunchKernelGGL(raster_kernel, dim3(HB * NTILE * NTILE), dim3(256), 0, stream,
                       pts, cnt, lst, idx_out, z_out, d2_out);
}